// GATLayer_31593779429804
// MI455X (gfx1250) — compile-verified
//
#include <hip/hip_runtime.h>
#include <hip/hip_bf16.h>

// out = x @ W^T  -- the reference's attention einsum contracts an index absent
// from h, so softmax rows sum to 1 and the whole attention path cancels.
// Full-precision f32 WMMA (V_WMMA_F32_16X16X4_F32), memory-bound problem.

typedef float v2f __attribute__((ext_vector_type(2)));
typedef float v8f __attribute__((ext_vector_type(8)));

#define IN_DIM   256
#define OUT_DIM  256
#define ROWS_TOT 8192   // B*N = 4*2048

// Block: 256 threads = 8 waves. Wave (wr, wc) computes a 16x64 tile:
//   rows [blockIdx.x*32 + wr*16, +16), cols [wc*64, +64) as 4 WMMA N-tiles.
__global__ __launch_bounds__(256)
void gat_h_gemm_wmma_f32(const float* __restrict__ X,
                         const float* __restrict__ W,
                         float* __restrict__ Out) {
  const int tid  = threadIdx.x;
  const int wave = tid >> 5;
  const int lane = tid & 31;
  const int lm   = lane & 15;   // M (for A) / N (for B,D) index within tile
  const int hi   = lane >> 4;   // 0: K={0,1}, 1: K={2,3} halves

  const int wr = wave >> 2;     // 0..1  (M sub-block of 16 rows)
  const int wc = wave & 3;      // 0..3  (N strip of 64 cols)

  const int m0 = blockIdx.x * 32 + wr * 16;
  const int n0 = wc * 64;

  // A fragment source: X[m0+lm][k + 2*hi .. +1]  -> contiguous b64 per lane
  const float* __restrict__ aptr = X + (size_t)(m0 + lm) * IN_DIM + 2 * hi;
  // B fragment source: B[k][n] = W[n][k]; lane supplies col n = n0+nt*16+lm,
  // rows k+2*hi..+1 -> contiguous b64 per lane within W's row.
  const float* __restrict__ bptr0 = W + (size_t)(n0 +  0 + lm) * IN_DIM + 2 * hi;
  const float* __restrict__ bptr1 = W + (size_t)(n0 + 16 + lm) * IN_DIM + 2 * hi;
  const float* __restrict__ bptr2 = W + (size_t)(n0 + 32 + lm) * IN_DIM + 2 * hi;
  const float* __restrict__ bptr3 = W + (size_t)(n0 + 48 + lm) * IN_DIM + 2 * hi;

  v8f acc0 = {};
  v8f acc1 = {};
  v8f acc2 = {};
  v8f acc3 = {};

#pragma unroll 4
  for (int k = 0; k < IN_DIM; k += 4) {
    v2f a  = *(const v2f*)(aptr  + k);
    v2f b0 = *(const v2f*)(bptr0 + k);
    v2f b1 = *(const v2f*)(bptr1 + k);
    v2f b2 = *(const v2f*)(bptr2 + k);
    v2f b3 = *(const v2f*)(bptr3 + k);

    // 8 args: (neg_a, A, neg_b, B, c_mod, C, reuse_a, reuse_b)
    acc0 = __builtin_amdgcn_wmma_f32_16x16x4_f32(false, a, false, b0,
                                                 (short)0, acc0, false, false);
    acc1 = __builtin_amdgcn_wmma_f32_16x16x4_f32(false, a, false, b1,
                                                 (short)0, acc1, false, false);
    acc2 = __builtin_amdgcn_wmma_f32_16x16x4_f32(false, a, false, b2,
                                                 (short)0, acc2, false, false);
    acc3 = __builtin_amdgcn_wmma_f32_16x16x4_f32(false, a, false, b3,
                                                 (short)0, acc3, false, false);
  }

  // D layout: VGPR r holds row (m0 + r + 8*hi), col n0 + nt*16 + lm.
  float* __restrict__ o = Out + (size_t)(m0 + 8 * hi) * OUT_DIM + n0 + lm;
#pragma unroll
  for (int r = 0; r < 8; ++r) {
    o[(size_t)r * OUT_DIM +  0] = acc0[r];
    o[(size_t)r * OUT_DIM + 16] = acc1[r];
    o[(size_t)r * OUT_DIM + 32] = acc2[r];
    o[(size_t)r * OUT_DIM + 48] = acc3[r];
  }
}

extern "C" void kernel_launch(void* const* d_in, const int* in_sizes, int n_in,
                              void* d_out, int out_size, void* d_ws, size_t ws_size,
                              hipStream_t stream) {
  // setup_inputs order: x, adj, W, a_w.  adj and a_w are mathematically dead
  // (attention weights sum to 1 over the contracted index).
  const float* X = (const float*)d_in[0];
  const float* W = (const float*)d_in[2];
  float* Out     = (float*)d_out;

  (void)in_sizes; (void)n_in; (void)out_size; (void)d_ws; (void)ws_size;

  dim3 grid(ROWS_TOT / 32, 1, 1);   // 256 blocks
  dim3 block(256, 1, 1);            // 8 waves
  gat_h_gemm_wmma_f32<<<grid, block, 0, stream>>>(X, W, Out);
}